// Model_83245056131580
// MI455X (gfx1250) — compile-verified
//
#include <hip/hip_runtime.h>
#include <hip/hip_bf16.h>
#include <cstdint>

// Problem constants (fixed by the reference)
#define N_NODES 64
#define T_STEPS 512
#define B_BATCH 64
#define P_WIN   32
#define RDIM    4096   // N*N rows of g / F
#define KDIM    4096   // reduction dim of the GEMM
#define TDIM    512    // columns of F
#define KC      32     // K chunk staged per LDS buffer

typedef __attribute__((ext_vector_type(2))) float v2f;
typedef __attribute__((ext_vector_type(8))) float v8f;

__device__ __forceinline__ v8f wmma_f32(v2f a, v2f b, v8f c) {
    return __builtin_amdgcn_wmma_f32_16x16x4_f32(false, a, false, b,
                                                 (short)0, c, false, false);
}

// Async DMA: global -> LDS, 16 bytes per lane, tracked by ASYNCcnt.
__device__ __forceinline__ void async_ld_b128(unsigned lds_byte_off, const float* gaddr) {
    asm volatile("global_load_async_to_lds_b128 %0, %1, off"
                 :: "v"(lds_byte_off), "v"(gaddr) : "memory");
}
__device__ __forceinline__ void wait_async0() {
    asm volatile("s_wait_asynccnt 0" ::: "memory");
}

// ---------------------------------------------------------------------------
// Kernel 0: w2 = weights^2 (elementwise, float4) so the GEMM B-operand can be
// DMA'd into LDS without a transform.
// ---------------------------------------------------------------------------
__global__ __launch_bounds__(256)
void square_kernel(const float4* __restrict__ wts, float4* __restrict__ w2) {
    int i = blockIdx.x * 256 + threadIdx.x;   // grid covers exactly 2M/4 vec4s
    float4 v = wts[i];
    v.x *= v.x; v.y *= v.y; v.z *= v.z; v.w *= v.w;
    w2[i] = v;
}

// ---------------------------------------------------------------------------
// Kernel 1: F = g @ w2   [4096x4096] x [4096x512] -> [4096x512]
// 128 threads = 4 waves. Block tile 64(M) x 64(T); each wave owns a 32x32
// tile as 2x2 WMMA accumulators (A/B fragments reused twice -> half the LDS
// read traffic per WMMA). K staged in chunks of 32 via double-buffered
// GLOBAL_LOAD_ASYNC_TO_LDS_B128 (ASYNCcnt), computed with
// V_WMMA_F32_16X16X4_F32 (precision-faithful f32 matrix path).
// ---------------------------------------------------------------------------
__global__ __launch_bounds__(128)
void gemm_f_kernel(const float* __restrict__ g,
                   const float* __restrict__ w2,
                   float* __restrict__ F) {
    // Strides chosen for 16B-aligned DMA writes AND conflict-free WMMA reads:
    //  sA stride 36: 36*4 % 16 == 0; {36*r mod 64 : r=0..15} all distinct.
    //  sB stride 72: row-to-row bank shift = 72*2 mod 64 = 16 -> the two
    //                half-waves (k, k+2) hit disjoint bank ranges.
    __shared__ float sA[2][64][36];   // [buf][m][k]
    __shared__ float sB[2][32][72];   // [buf][k][t]

    const int tid  = threadIdx.x;
    const int lane = tid & 31;
    const int wave = tid >> 5;
    const int r0   = blockIdx.x * 64;       // M tile base
    const int t0   = blockIdx.y * 64;       // T tile base
    const int wm   = (wave & 1) * 32;       // wave's 32-row half
    const int wn   = (wave >> 1) * 32;      // wave's 32-col half
    const int ml   = lane & 15;             // M (A) / N (B,C) within 16x16
    const int kp   = (lane >> 4) << 1;      // K-pair select per half-wave

    v8f acc00 = {}, acc01 = {}, acc10 = {}, acc11 = {};

    auto stage = [&](int k0, int buf) {
        // A: 64 rows x 32 k = 512 x B128; 4 per thread. Coalesced along k.
        #pragma unroll
        for (int it = 0; it < 4; ++it) {
            int idx = tid + it * 128;
            int row = idx >> 3;
            int kk  = (idx & 7) << 2;
            unsigned lds = (unsigned)(uintptr_t)&sA[buf][row][kk];
            async_ld_b128(lds, g + (size_t)(r0 + row) * KDIM + (k0 + kk));
        }
        // B: 32 k x 64 t = 512 x B128; 4 per thread. Coalesced along t.
        #pragma unroll
        for (int it = 0; it < 4; ++it) {
            int idx = tid + it * 128;
            int kk  = idx >> 4;
            int t   = (idx & 15) << 2;
            unsigned lds = (unsigned)(uintptr_t)&sB[buf][kk][t];
            async_ld_b128(lds, w2 + (size_t)(k0 + kk) * TDIM + (t0 + t));
        }
    };

    auto compute = [&](int buf) {
        #pragma unroll
        for (int kk = 0; kk < KC; kk += 4) {
            v2f a0, a1, b0, b1;
            a0.x = sA[buf][wm + ml][kk + kp];
            a0.y = sA[buf][wm + ml][kk + kp + 1];
            a1.x = sA[buf][wm + 16 + ml][kk + kp];
            a1.y = sA[buf][wm + 16 + ml][kk + kp + 1];
            b0.x = sB[buf][kk + kp][wn + ml];
            b0.y = sB[buf][kk + kp + 1][wn + ml];
            b1.x = sB[buf][kk + kp][wn + 16 + ml];
            b1.y = sB[buf][kk + kp + 1][wn + 16 + ml];
            acc00 = wmma_f32(a0, b0, acc00);
            acc01 = wmma_f32(a0, b1, acc01);
            acc10 = wmma_f32(a1, b0, acc10);
            acc11 = wmma_f32(a1, b1, acc11);
        }
    };

    stage(0, 0);
    wait_async0();
    __syncthreads();

    const int nchunks = KDIM / KC;   // 128
    for (int c = 1; c < nchunks; ++c) {
        stage(c * KC, c & 1);        // DMA next chunk into the idle buffer
        compute((c - 1) & 1);        // math on the ready buffer meanwhile
        wait_async0();               // my DMAs done ...
        __syncthreads();             // ... everyone's DMAs visible / buffer free
    }
    compute((nchunks - 1) & 1);

    // C/D layout: VGPR v -> M = v + 8*(lane>=16), N = lane&15
    const int rb = r0 + wm + ((lane >> 4) << 3);
    const int cb = t0 + wn + ml;
    #pragma unroll
    for (int v = 0; v < 8; ++v) {
        F[(size_t)(rb + v) * TDIM + cb]            = acc00[v];
        F[(size_t)(rb + v) * TDIM + cb + 16]       = acc01[v];
        F[(size_t)(rb + 16 + v) * TDIM + cb]       = acc10[v];
        F[(size_t)(rb + 16 + v) * TDIM + cb + 16]  = acc11[v];
    }
}

// ---------------------------------------------------------------------------
// Kernel 2: w[t,n,m] = softmax_m( F[n*64+m, t] )
// grid: (T/32, N). Stage 64(m) x 32(t) panel of F in LDS (coalesced along t),
// softmax over m per t, write w coalesced along m.
// ---------------------------------------------------------------------------
__global__ __launch_bounds__(256)
void softmax_kernel(const float* __restrict__ F, float* __restrict__ w) {
    __shared__ float sF[64][33];
    __shared__ float smx[32];
    __shared__ float srs[32];

    const int tid = threadIdx.x;
    const int t0  = blockIdx.x * 32;
    const int n   = blockIdx.y;

    #pragma unroll
    for (int i = 0; i < 8; ++i) {
        int idx = tid + i * 256;
        int m   = idx >> 5;
        int tl  = idx & 31;
        sF[m][tl] = F[(size_t)(n * 64 + m) * TDIM + (t0 + tl)];
    }
    __syncthreads();

    if (tid < 32) {
        float mx = -3.402823466e38f;
        #pragma unroll
        for (int m = 0; m < 64; ++m) mx = fmaxf(mx, sF[m][tid]);
        float s = 0.f;
        #pragma unroll
        for (int m = 0; m < 64; ++m) s += __expf(sF[m][tid] - mx);
        smx[tid] = mx;
        srs[tid] = 1.f / s;
    }
    __syncthreads();

    #pragma unroll
    for (int i = 0; i < 8; ++i) {
        int idx = tid + i * 256;
        int m   = idx & 63;     // consecutive tid -> consecutive m -> coalesced
        int tl  = idx >> 6;
        w[((size_t)(t0 + tl) * 64 + n) * 64 + m] =
            __expf(sF[m][tl] - smx[tl]) * srs[tl];
    }
}

// ---------------------------------------------------------------------------
// Kernel 3: Z[b,n] = sum_{p,m} w[x_i[b,p], n, m] * x[b, m, p]
// One block per batch; x[b] tile + indices cached in LDS; 4 lanes per n,
// shfl_down reduction (wave32).
// ---------------------------------------------------------------------------
__global__ __launch_bounds__(256)
void gather_z_kernel(const float* __restrict__ x, const int* __restrict__ xi,
                     const float* __restrict__ w, float* __restrict__ Z) {
    __shared__ float sx[64][33];   // [m][p]
    __shared__ int   sidx[32];

    const int tid = threadIdx.x;
    const int b   = blockIdx.x;

    #pragma unroll
    for (int i = 0; i < 8; ++i) {
        int idx = tid + i * 256;
        int m   = idx >> 5;
        int p   = idx & 31;
        sx[m][p] = x[(size_t)b * (N_NODES * P_WIN) + m * P_WIN + p];
    }
    if (tid < 32) sidx[tid] = xi[b * P_WIN + tid];
    __syncthreads();

    const int n = tid >> 2;   // 64 outputs, 4 threads each
    const int q = tid & 3;    // each handles 8 of 32 time steps
    float acc = 0.f;
    #pragma unroll
    for (int pp = 0; pp < 8; ++pp) {
        int p = q * 8 + pp;
        int t = sidx[p];
        const float* wrow = w + ((size_t)t * 64 + n) * 64;
        #pragma unroll
        for (int m = 0; m < 64; ++m) acc += wrow[m] * sx[m][p];
    }
    acc += __shfl_down(acc, 2);
    acc += __shfl_down(acc, 1);
    if (q == 0) Z[b * N_NODES + n] = acc;
}

// ---------------------------------------------------------------------------
extern "C" void kernel_launch(void* const* d_in, const int* in_sizes, int n_in,
                              void* d_out, int out_size, void* d_ws, size_t ws_size,
                              hipStream_t stream) {
    const float* x   = (const float*)d_in[0];   // [B, N, P]
    const int*   xi  = (const int*)  d_in[1];   // [B, P]
    const float* g   = (const float*)d_in[2];   // [N*N, N*N]
    const float* wts = (const float*)d_in[3];   // [N*N, T]

    float* Z = (float*)d_out;                      // [B, N]
    float* F = (float*)d_out + B_BATCH * N_NODES;  // [N*N, T]

    // Workspace: w2 (weights^2, GEMM input) and w (softmax output) have
    // disjoint lifetimes -> alias both onto the same 8 MB region.
    float* w2 = (float*)d_ws;
    float* w  = (float*)d_ws;

    square_kernel<<<RDIM * TDIM / 4 / 256, 256, 0, stream>>>(
        (const float4*)wts, (float4*)w2);
    gemm_f_kernel<<<dim3(RDIM / 64, TDIM / 64), 128, 0, stream>>>(g, w2, F);
    softmax_kernel<<<dim3(TDIM / 32, N_NODES), 256, 0, stream>>>(F, w);
    gather_z_kernel<<<B_BATCH, 256, 0, stream>>>(x, xi, w, Z);
}